// Ses2Seq_38732015075468
// MI455X (gfx1250) — compile-verified
//
#include <hip/hip_runtime.h>
#include <math.h>

// Problem constants (from reference)
#define L     880
#define HID   8
#define OUT   96
#define DS    48
#define T     64
#define BATCH 4
#define NCONV 63          // circular-conv length = T-1
#define CW    64          // chain width: cols 0..47 = A^t B, 48 = A^t theta0, 49..52 = A^t sig1[b]
#define CSTEP (L * CW)    // floats per chain timestep

typedef float v2f __attribute__((ext_vector_type(2)));
typedef float v8f __attribute__((ext_vector_type(8)));

// ---------------------------------------------------------------------------
// Generic fp32 WMMA matmul: D[z] = Am(880x880) @ Bm[z](880 x ncols)
//   grid.x = row tiles (55), grid.y = col-tile groups of 4, grid.z = chains
//   block = 128 threads = 4 waves; each wave owns one 16x16 output tile.
//   Col-tile guard is wave-uniform so EXEC is all-ones around WMMA.
//   LDB/LDD are compile-time so B-operand addresses fold to constant offsets.
// ---------------------------------------------------------------------------
template <int LDB, int LDD>
__global__ void wmma_mm_kernel(const float* __restrict__ Am,
                               const float* __restrict__ Bb, long bz,
                               float* __restrict__ Db, long dz,
                               int ncols) {
    const int lane = threadIdx.x & 31;
    const int jt   = blockIdx.y * 4 + (threadIdx.x >> 5);   // column tile
    if (jt * 16 >= ncols) return;                           // wave-uniform exit

    const int itile = blockIdx.x;
    const float* Bm = Bb + (size_t)blockIdx.z * bz;
    float*       Dm = Db + (size_t)blockIdx.z * dz;

    const int mrow  = lane & 15;                 // A-frag: M = lane%16
    const int khalf = (lane >> 4) << 1;          // K sub-offset: 0 or 2
    const int col   = jt * 16 + (lane & 15);     // B/D-frag: N

    const float* Arow = Am + (size_t)(itile * 16 + mrow) * L;

    v8f acc = {};
    for (int k = 0; k < L; k += 4) {
        v2f a, b;
        a.x = Arow[k + khalf + 0];
        a.y = Arow[k + khalf + 1];
        b.x = Bm[(size_t)(k + khalf + 0) * LDB + col];
        b.y = Bm[(size_t)(k + khalf + 1) * LDB + col];
        acc = __builtin_amdgcn_wmma_f32_16x16x4_f32(false, a, false, b,
                                                    (short)0, acc, false, false);
    }

    const int rbase = itile * 16 + ((lane >> 4) << 3);   // D-frag: M = v + 8*(lane>=16)
    #pragma unroll
    for (int v = 0; v < 8; ++v)
        Dm[(size_t)(rbase + v) * LDD + col] = acc[v];
}

// ---------------------------------------------------------------------------
// Stage 0: build C_0 (880 x 64): [B | theta0 | -B@xs[b,0] (x4) | zeros]
// also seed thetas[b][0][:] = theta0
// ---------------------------------------------------------------------------
__global__ void ses2seq_init(const float* __restrict__ Bm,
                             const float* __restrict__ theta0,
                             const float* __restrict__ xs,
                             float* __restrict__ C0,
                             float* __restrict__ thetas) {
    int l = blockIdx.x * blockDim.x + threadIdx.x;
    if (l >= L) return;
    float* row = C0 + (size_t)l * CW;
    const float* brow = Bm + (size_t)l * DS;
    #pragma unroll 4
    for (int d = 0; d < DS; ++d) row[d] = brow[d];
    row[DS] = theta0[l];
    #pragma unroll
    for (int b = 0; b < BATCH; ++b) {
        float acc = 0.f;
        const float* x0 = xs + (size_t)b * T * DS;   // xs[b,0,:]
        for (int d = 0; d < DS; ++d) acc += brow[d] * x0[d];
        row[DS + 1 + b] = -acc;                      // sig1[b][l]
    }
    for (int c = DS + 1 + BATCH; c < CW; ++c) row[c] = 0.f;
    #pragma unroll
    for (int b = 0; b < BATCH; ++b)
        thetas[((size_t)b * T + 0) * L + l] = theta0[l];
}

// ---------------------------------------------------------------------------
// Stage 2: circular conv (length 63) + phi0, assembled directly into thetas.
//   thetas[b][tc+1][l] = sum_{m=0..62} K_conv[m][l,:] . u[b,(tc-m)%63,:]
//                        + C[tc][l,48] + C[(tc-1)%63][l,49+b]
//   K_conv[m] = C[m][:, :48] - (m>0 ? C[m-1][:, :48] : 0)   (folded into A-frag)
// grid: (55 l-tiles, 4 batches), block: 128 = 4 waves (one 16-wide tc tile each)
// ---------------------------------------------------------------------------
__global__ void ses2seq_conv(const float* __restrict__ C,
                             const float* __restrict__ xs,
                             float* __restrict__ thetas) {
    const int lane  = threadIdx.x & 31;
    const int wv    = threadIdx.x >> 5;          // tc tile 0..3
    const int ltile = blockIdx.x;                // 0..54
    const int b     = blockIdx.y;                // 0..3
    const int khalf = (lane >> 4) << 1;
    const int lrow  = ltile * 16 + (lane & 15);  // A-frag M
    const int tc    = wv * 16 + (lane & 15);     // B/D-frag N; tc==63 invalid
    const int tcc   = (tc >= NCONV) ? 0 : tc;    // safe index for gathers

    const float* u = xs + (size_t)b * T * DS;    // u[s][d] = xs[b][s+1][d]

    v8f acc = {};
    for (int m = 0; m < NCONV; ++m) {
        const float* Km = C + (size_t)m * CSTEP;
        const float* Kp = C + (size_t)(m > 0 ? m - 1 : 0) * CSTEP;
        const float  w  = (m > 0) ? 1.f : 0.f;
        int s = tcc - m; if (s < 0) s += NCONV;              // (tc-m) mod 63
        const float* urow = u + (size_t)(s + 1) * DS;        // xs[b][s+1][:]
        #pragma unroll 4
        for (int d = 0; d < DS; d += 4) {
            const int dk = d + khalf;
            v2f a, bb;
            a.x = Km[(size_t)lrow * CW + dk + 0] - w * Kp[(size_t)lrow * CW + dk + 0];
            a.y = Km[(size_t)lrow * CW + dk + 1] - w * Kp[(size_t)lrow * CW + dk + 1];
            bb.x = urow[dk + 0];
            bb.y = urow[dk + 1];
            acc = __builtin_amdgcn_wmma_f32_16x16x4_f32(false, a, false, bb,
                                                        (short)0, acc, false, false);
        }
    }

    if (tc < NCONV) {
        const int l0 = ltile * 16 + ((lane >> 4) << 3);
        const int zc = (tc - 1 + NCONV) % NCONV;
        #pragma unroll
        for (int v = 0; v < 8; ++v) {
            const int l = l0 + v;
            float val = acc[v]
                      + C[(size_t)tc * CSTEP + (size_t)l * CW + DS]            // A^tc theta0
                      + C[(size_t)zc * CSTEP + (size_t)l * CW + DS + 1 + b];   // A^zc sig1[b]
            thetas[((size_t)b * T + (tc + 1)) * L + l] = val;
        }
    }
}

// ---------------------------------------------------------------------------
// Stage 3: per-(b,t) MLP head. 256 blocks x 96 threads.
// ---------------------------------------------------------------------------
__global__ void ses2seq_head(const float* __restrict__ thetas,
                             const float* __restrict__ ts,
                             float* __restrict__ out) {
    const int bt = blockIdx.x;                   // b*64 + t
    const float* th = thetas + (size_t)bt * L;
    __shared__ float h[HID];
    if (threadIdx.x < HID) {
        float x = th[threadIdx.x] * ts[bt] + th[HID + threadIdx.x];
        h[threadIdx.x] = x > 0.f ? x : 0.f;      // relu(W0*t + b0)
    }
    __syncthreads();
    const int o = threadIdx.x;                   // 0..95
    float acc = th[2 * HID + OUT * HID + o];     // b1[o]
    const float* w1 = th + 2 * HID + o * HID;    // W1[o][:]
    #pragma unroll
    for (int k = 0; k < HID; ++k) acc += w1[k] * h[k];
    float r;
    if (o < DS) {
        r = tanhf(acc);
    } else {                                     // stable softplus = log1p(exp(x))
        r = acc > 0.f ? acc + log1pf(expf(-acc)) : log1pf(expf(acc));
    }
    out[(size_t)bt * OUT + o] = r;
}

// ---------------------------------------------------------------------------
extern "C" void kernel_launch(void* const* d_in, const int* in_sizes, int n_in,
                              void* d_out, int out_size, void* d_ws, size_t ws_size,
                              hipStream_t stream) {
    const float* xs     = (const float*)d_in[0];   // (4,64,48)
    const float* ts     = (const float*)d_in[1];   // (4,64)
    const float* theta0 = (const float*)d_in[2];   // (880,)
    const float* A      = (const float*)d_in[3];   // (880,880)
    const float* Bm     = (const float*)d_in[4];   // (880,48)
    (void)in_sizes; (void)n_in; (void)out_size; (void)ws_size;

    float* A2     = (float*)d_ws;                        // 880*880
    float* A4     = A2 + (size_t)L * L;                  // 880*880
    float* C      = A4 + (size_t)L * L;                  // 63 * 880 * 64
    float* thetas = C + (size_t)NCONV * CSTEP;           // 4 * 64 * 880
    float* out    = (float*)d_out;                       // (4,64,96)

    // Stage 0: build C_0 and thetas t=0 rows
    ses2seq_init<<<(L + 255) / 256, 256, 0, stream>>>(Bm, theta0, xs, C, thetas);

    // Power doubling: A2 = A@A, A4 = A2@A2 (fully parallel, cuts serial depth 62 -> ~18)
    wmma_mm_kernel<L, L><<<dim3(55, 14, 1), 128, 0, stream>>>(A,  A,  0, A2, 0, L);
    wmma_mm_kernel<L, L><<<dim3(55, 14, 1), 128, 0, stream>>>(A2, A2, 0, A4, 0, L);

    // Seed the four interleaved chains:
    //   C1 = A @ C0 ; (C2,C3) = A2 @ (C0,C1)
    wmma_mm_kernel<CW, CW><<<dim3(55, 1, 1), 128, 0, stream>>>(
        A, C, 0, C + CSTEP, 0, CW);
    wmma_mm_kernel<CW, CW><<<dim3(55, 1, 2), 128, 0, stream>>>(
        A2, C, CSTEP, C + (size_t)2 * CSTEP, CSTEP, CW);

    // Quad steps: (C_{4j}..C_{4j+3}) = A4 @ (C_{4j-4}..C_{4j-1}); stop at C_62
    for (int j = 1; j <= 15; ++j) {
        int nz = (j == 15) ? 3 : 4;              // last launch: only C60..C62
        wmma_mm_kernel<CW, CW><<<dim3(55, 1, nz), 128, 0, stream>>>(
            A4,
            C + (size_t)(4 * j - 4) * CSTEP, CSTEP,
            C + (size_t)(4 * j) * CSTEP, CSTEP, CW);
    }

    // Stage 2: circular conv + phi0 -> thetas[:,1:,:]
    ses2seq_conv<<<dim3(55, BATCH), 128, 0, stream>>>(C, xs, thetas);

    // Stage 3: MLP head -> output
    ses2seq_head<<<BATCH * T, OUT, 0, stream>>>(thetas, ts, out);
}